// AssociativeMemoryBlock_78932908966648
// MI455X (gfx1250) — compile-verified
//
#include <hip/hip_runtime.h>

typedef __attribute__((ext_vector_type(2))) float v2f;
typedef __attribute__((ext_vector_type(8))) float v8f;

#define Bc 4
#define Sc 4096
#define Ec 512
#define Dc 64
#define Rc 2
#define NSLOT 7
#define CHUNK 64
#define NCHUNK (Sc / CHUNK)
#define EPS 1e-12f

static __device__ __forceinline__ v8f wmma_f32_16x16x4(v2f a, v2f b, v8f c) {
  // D = A(16x4,f32) * B(4x16,f32) + C(16x16,f32)
  return __builtin_amdgcn_wmma_f32_16x16x4_f32(false, a, false, b, (short)0, c,
                                               false, false);
}

// --- CDNA5 async DMA helpers (ASYNCcnt-tracked, per CDNA5 ISA) ----------
// lds_off: byte offset into the workgroup LDS allocation (dynamic LDS base=0)
static __device__ __forceinline__ void async_g2l_b128(unsigned lds_off,
                                                      const void* gptr) {
  asm volatile("global_load_async_to_lds_b128 %0, %1, off"
               :
               : "v"(lds_off), "v"(gptr)
               : "memory");
}
static __device__ __forceinline__ void async_l2g_b128(void* gptr,
                                                      unsigned lds_off) {
  asm volatile("global_store_async_from_lds_b128 %0, %1, off"
               :
               : "v"(gptr), "v"(lds_off)
               : "memory");
}
static __device__ __forceinline__ void wait_async0() {
  asm volatile("s_wait_asynccnt 0x0" ::: "memory");
}

// ---------------------------------------------------------------------------
// Kernel 1: V = x@Wv^T ; K = l2norm(x@Wk^T) ; Qr0 = l2norm(V @ Q[r])
// Block = 128 threads (4 waves) handles 16 rows of x.
// LDS arena (floats): XS[16*512] | VS[16*64] | KS[16*64] | QS[2*16*64] | NRM[32]
// ---------------------------------------------------------------------------
#define P_XS 0
#define P_VS 8192
#define P_KS 9216
#define P_QS 10240
#define P_NRM 12288
#define P_FLOATS 12320

__global__ __launch_bounds__(128)
void proj_kernel(const float* __restrict__ x, const float* __restrict__ Wv,
                 const float* __restrict__ Wk, const float* __restrict__ Qm,
                 float* __restrict__ V, float* __restrict__ K,
                 float* __restrict__ Qr0) {
  extern __shared__ float lds[];
  const int tid = threadIdx.x;
  const int lane = tid & 31;
  const int wave = tid >> 5;
  const long row0 = (long)blockIdx.x * 16;  // global row in [0, B*S)

  // async-stage 16 rows of x (x rows are contiguous -> flat copy), 16B/lane
  for (int i = tid * 4; i < 16 * Ec; i += 128 * 4)
    async_g2l_b128((unsigned)((P_XS + i) * 4), &x[row0 * Ec + i]);
  wait_async0();
  __syncthreads();

  const int arow = lane & 15;         // A-row held by this lane
  const int kb = (lane >> 4) << 1;    // K sub-offset {0,2}
  const int bn = lane & 15;           // B/D column held by this lane
  const int mofs = (lane >> 4) << 3;  // D-row offset {0,8}

  // V (p=0) and raw-K (p=1) projections: combo = p*4 + coltile
  for (int cmb = wave; cmb < 8; cmb += 4) {
    const int p = cmb >> 2;
    const int ct = cmb & 3;
    const float* Wb = (p == 0 ? Wv : Wk) + (long)(ct * 16 + bn) * Ec;
    v8f acc = {};
    for (int kk = 0; kk < Ec; kk += 4) {
      v2f a, b;
      a.x = lds[P_XS + arow * Ec + kk + kb];
      a.y = lds[P_XS + arow * Ec + kk + kb + 1];
      b.x = Wb[kk + kb];
      b.y = Wb[kk + kb + 1];
      acc = wmma_f32_16x16x4(a, b, acc);
    }
    float* dstT = lds + (p == 0 ? P_VS : P_KS);
#pragma unroll
    for (int i = 0; i < 8; ++i)
      dstT[(i + mofs) * Dc + ct * 16 + bn] = acc[i];
  }
  __syncthreads();

  // l2-normalize K rows, publish V and K
  if (tid < 16) {
    float s = 0.f;
    for (int j = 0; j < Dc; ++j) {
      float kv = lds[P_KS + tid * Dc + j];
      s += kv * kv;
    }
    lds[P_NRM + tid] = fmaxf(sqrtf(s), EPS);
  }
  __syncthreads();
  for (int i = tid; i < 16 * Dc; i += 128) {
    int r = i >> 6, j = i & 63;
    K[(row0 + r) * Dc + j] = lds[P_KS + i] / lds[P_NRM + r];
    V[(row0 + r) * Dc + j] = lds[P_VS + i];
  }
  __syncthreads();

  // Qr0: Qhat[row][q] = sum_v vs[row][v] * Q[r][v][q] ; combo = r*4 + coltile
  for (int cmb = wave; cmb < 8; cmb += 4) {
    const int r = cmb >> 2;
    const int ct = cmb & 3;
    const float* Qb = Qm + (long)r * Dc * Dc;
    v8f acc = {};
    for (int kk = 0; kk < Dc; kk += 4) {
      v2f a, b;
      a.x = lds[P_VS + arow * Dc + kk + kb];
      a.y = lds[P_VS + arow * Dc + kk + kb + 1];
      b.x = Qb[(kk + kb) * Dc + ct * 16 + bn];
      b.y = Qb[(kk + kb + 1) * Dc + ct * 16 + bn];
      acc = wmma_f32_16x16x4(a, b, acc);
    }
#pragma unroll
    for (int i = 0; i < 8; ++i)
      lds[P_QS + r * 1024 + (i + mofs) * Dc + ct * 16 + bn] = acc[i];
  }
  __syncthreads();

  if (tid < 32) {
    float s = 0.f;
    for (int j = 0; j < Dc; ++j) {
      float qv = lds[P_QS + tid * Dc + j];
      s += qv * qv;
    }
    lds[P_NRM + tid] = fmaxf(sqrtf(s), EPS);
  }
  __syncthreads();
  for (int i = tid; i < Rc * 16 * Dc; i += 128) {
    int r = i >> 10, rw = (i >> 6) & 15, j = i & 63;
    Qr0[((row0 + rw) * Rc + r) * Dc + j] =
        lds[P_QS + i] / lds[P_NRM + r * 16 + rw];
  }
}

// ---------------------------------------------------------------------------
// Kernel 2: sequential delta-rule scan; async-spill M at chunk boundaries.
// One block per batch; M lives in LDS.
// LDS arena (floats): M[64*64] | KK[64] | VV[64] | MK[64]
// ---------------------------------------------------------------------------
#define B_M 0
#define B_KK 4096
#define B_VV 4160
#define B_MK 4224
#define B_FLOATS 4288

__global__ __launch_bounds__(256)
void boundary_kernel(const float* __restrict__ M0, const float* __restrict__ K,
                     const float* __restrict__ V, float* __restrict__ Mb) {
  extern __shared__ float lds[];
  const int b = blockIdx.x;
  const int tid = threadIdx.x;
  for (int i = tid * 4; i < Dc * Dc; i += 256 * 4)
    async_g2l_b128((unsigned)((B_M + i) * 4), &M0[(long)b * Dc * Dc + i]);
  wait_async0();
  __syncthreads();

  const int row = tid >> 2;
  const int part = tid & 3;
  for (int t = 0; t < Sc; ++t) {
    if ((t & (CHUNK - 1)) == 0) {
      const long base = ((long)b * NCHUNK + (t >> 6)) * (long)Dc * Dc;
      for (int i = tid * 4; i < Dc * Dc; i += 256 * 4)
        async_l2g_b128(&Mb[base + i], (unsigned)((B_M + i) * 4));
      wait_async0();  // LDS reads must finish before this step's M update
    }
    const long rr = ((long)b * Sc + t) * Dc;
    if (tid < 32) {  // async-stage k_t and v_t (16B per lane)
      unsigned loff;
      const float* g;
      if (tid < 16) {
        loff = (unsigned)((B_KK + tid * 4) * 4);
        g = &K[rr + tid * 4];
      } else {
        loff = (unsigned)((B_VV + (tid - 16) * 4) * 4);
        g = &V[rr + (tid - 16) * 4];
      }
      async_g2l_b128(loff, g);
    }
    wait_async0();
    __syncthreads();
    float p = 0.f;
#pragma unroll
    for (int j = 0; j < 16; ++j)
      p += lds[B_M + row * Dc + part * 16 + j] * lds[B_KK + part * 16 + j];
    p += __shfl_xor(p, 1);
    p += __shfl_xor(p, 2);
    if (part == 0) lds[B_MK + row] = p;
    __syncthreads();
    const float coef = lds[B_VV + row] - lds[B_MK + row];
#pragma unroll
    for (int j = 0; j < 16; ++j)
      lds[B_M + row * Dc + part * 16 + j] += coef * lds[B_KK + part * 16 + j];
    __syncthreads();
  }
}

// ---------------------------------------------------------------------------
// Kernel 3: per-chunk replay + fused 2-hop retrieval (no M_all in HBM).
// One block per (batch, chunk).
// LDS (floats): M[4096] | QMAT[2*4096] | KK[64] | VV[64] | MK[64] |
//               Q[6*64] | VR[7*64] | NRM[4]
// ---------------------------------------------------------------------------
#define C_M 0
#define C_QM 4096
#define C_KK 12288
#define C_VV 12352
#define C_MK 12416
#define C_Q 12480
#define C_VR 12864
#define C_NRM 13312
#define C_FLOATS 13316

__global__ __launch_bounds__(256)
void chunk_kernel(const float* __restrict__ Mb, const float* __restrict__ K,
                  const float* __restrict__ V, const float* __restrict__ Qr0,
                  const float* __restrict__ Qm, float* __restrict__ Vret,
                  float* __restrict__ out_tail) {
  extern __shared__ float lds[];
  const int b = blockIdx.x / NCHUNK;
  const int c = blockIdx.x % NCHUNK;
  const int tid = threadIdx.x;

  for (int i = tid * 4; i < Dc * Dc; i += 256 * 4)
    async_g2l_b128((unsigned)((C_M + i) * 4),
                   &Mb[((long)b * NCHUNK + c) * (long)Dc * Dc + i]);
  for (int i = tid * 4; i < Rc * Dc * Dc; i += 256 * 4)
    async_g2l_b128((unsigned)((C_QM + i) * 4), &Qm[i]);
  wait_async0();
  __syncthreads();

  const int row4 = tid >> 2, part4 = tid & 3;

  for (int tt = 0; tt < CHUNK; ++tt) {
    const long t = (long)c * CHUNK + tt;
    const long rr = (long)b * Sc + t;
    if (tid < 64) {  // async-stage k_t, v_t and both hop-1 queries
      unsigned loff;
      const float* g;
      if (tid < 16) {
        loff = (unsigned)((C_KK + tid * 4) * 4);
        g = &K[rr * Dc + tid * 4];
      } else if (tid < 32) {
        loff = (unsigned)((C_VV + (tid - 16) * 4) * 4);
        g = &V[rr * Dc + (tid - 16) * 4];
      } else {
        loff = (unsigned)((C_Q + (tid - 32) * 4) * 4);
        g = &Qr0[rr * Rc * Dc + (tid - 32) * 4];  // q[0],q[1] contiguous
      }
      async_g2l_b128(loff, g);
    }
    wait_async0();
    __syncthreads();

    // recurrence step -> M_t
    float p = 0.f;
#pragma unroll
    for (int j = 0; j < 16; ++j)
      p += lds[C_M + row4 * Dc + part4 * 16 + j] * lds[C_KK + part4 * 16 + j];
    p += __shfl_xor(p, 1);
    p += __shfl_xor(p, 2);
    if (part4 == 0) lds[C_MK + row4] = p;
    __syncthreads();
    const float coef = lds[C_VV + row4] - lds[C_MK + row4];
#pragma unroll
    for (int j = 0; j < 16; ++j)
      lds[C_M + row4 * Dc + part4 * 16 + j] += coef * lds[C_KK + part4 * 16 + j];
    __syncthreads();

    // hop 1: vr[1+n] = M_t @ q[n]
    {
      const int n = tid >> 7, rw = (tid >> 1) & 63, pp = tid & 1;
      float s = 0.f;
#pragma unroll
      for (int j = 0; j < 32; ++j)
        s += lds[C_M + rw * Dc + pp * 32 + j] * lds[C_Q + n * Dc + pp * 32 + j];
      s += __shfl_xor(s, 1);
      if (pp == 0) lds[C_VR + (1 + n) * Dc + rw] = s;
    }
    __syncthreads();

    // hop-2 queries: q[2+m][qq] = sum_v vr[1+n][v] * Q[r'][v][qq]
    {
      const int m = tid >> 6, qq = tid & 63;
      const int n = m >> 1, rp = m & 1;
      float s = 0.f;
      for (int v = 0; v < Dc; ++v)
        s += lds[C_VR + (1 + n) * Dc + v] * lds[C_QM + rp * 4096 + v * Dc + qq];
      lds[C_Q + (2 + m) * Dc + qq] = s;
    }
    __syncthreads();
    if (tid < 4) {
      float s = 0.f;
      for (int j = 0; j < Dc; ++j) {
        float qv = lds[C_Q + (2 + tid) * Dc + j];
        s += qv * qv;
      }
      lds[C_NRM + tid] = fmaxf(sqrtf(s), EPS);
    }
    __syncthreads();
    {
      const int m = tid >> 6, qq = tid & 63;
      lds[C_Q + (2 + m) * Dc + qq] /= lds[C_NRM + m];
    }
    __syncthreads();

    // hop 2: vr[3+m] = M_t @ q[2+m]
    {
      const int m = tid >> 6, rw = tid & 63;
      float s = 0.f;
      for (int j = 0; j < Dc; ++j)
        s += lds[C_M + rw * Dc + j] * lds[C_Q + (2 + m) * Dc + j];
      lds[C_VR + (3 + m) * Dc + rw] = s;
    }
    __syncthreads();

    // emit the 7 retrieval slots
    for (int i = tid; i < NSLOT * Dc; i += 256) {
      const int slot = i >> 6, j = i & 63;
      const float val = (slot == 0) ? lds[C_VV + j] : lds[C_VR + i];
      Vret[(rr * NSLOT + slot) * Dc + j] = val;
    }
    if (t == Sc - 1) {
      for (int i = tid * 4; i < Dc * Dc; i += 256 * 4)
        async_l2g_b128(&out_tail[(long)b * Dc * Dc + i],
                       (unsigned)((C_M + i) * 4));
      wait_async0();
    }
    __syncthreads();
  }
}

// ---------------------------------------------------------------------------
// Kernel 4: out = V_ret @ Wout^T  (M=B*S*7, N=E, K=D) — one 16x16 tile/wave.
// ---------------------------------------------------------------------------
__global__ __launch_bounds__(256)
void out_gemm_kernel(const float* __restrict__ Vret,
                     const float* __restrict__ Wout, float* __restrict__ out) {
  const int lane = threadIdx.x & 31;
  const int wave = threadIdx.x >> 5;
  const long tile = (long)blockIdx.x * 8 + wave;
  const long rowT = tile / (Ec / 16);
  const int colT = (int)(tile % (Ec / 16));
  const long row0 = rowT * 16;
  const int arow = lane & 15;
  const int kb = (lane >> 4) << 1;
  const int bn = lane & 15;
  const int mofs = (lane >> 4) << 3;

  const float* A = Vret + (row0 + arow) * Dc;
  const float* Bp = Wout + (long)(colT * 16 + bn) * Dc;
  v8f acc = {};
#pragma unroll
  for (int kk = 0; kk < Dc; kk += 4) {
    v2f a, b;
    a.x = A[kk + kb];
    a.y = A[kk + kb + 1];
    b.x = Bp[kk + kb];
    b.y = Bp[kk + kb + 1];
    acc = wmma_f32_16x16x4(a, b, acc);
  }
#pragma unroll
  for (int i = 0; i < 8; ++i)
    out[(row0 + i + mofs) * Ec + colT * 16 + bn] = acc[i];
}

// ---------------------------------------------------------------------------
extern "C" void kernel_launch(void* const* d_in, const int* in_sizes, int n_in,
                              void* d_out, int out_size, void* d_ws,
                              size_t ws_size, hipStream_t stream) {
  const float* x = (const float*)d_in[0];
  const float* M0 = (const float*)d_in[1];
  const float* Wv = (const float*)d_in[2];
  const float* Qm = (const float*)d_in[3];
  const float* Wk = (const float*)d_in[4];
  const float* Wout = (const float*)d_in[5];
  float* out = (float*)d_out;

  float* V = (float*)d_ws;                           // B*S*D
  float* K = V + (size_t)Bc * Sc * Dc;               // B*S*D
  float* Qr0 = K + (size_t)Bc * Sc * Dc;             // B*S*R*D
  float* Mb = Qr0 + (size_t)Bc * Sc * Rc * Dc;       // B*NCHUNK*D*D
  float* Vret = Mb + (size_t)Bc * NCHUNK * Dc * Dc;  // B*S*7*D

  proj_kernel<<<(Bc * Sc) / 16, 128, P_FLOATS * 4, stream>>>(x, Wv, Wk, Qm, V,
                                                             K, Qr0);
  boundary_kernel<<<Bc, 256, B_FLOATS * 4, stream>>>(M0, K, V, Mb);
  chunk_kernel<<<Bc * NCHUNK, 256, C_FLOATS * 4, stream>>>(
      Mb, K, V, Qr0, Qm, Vret, out + (size_t)Bc * Sc * NSLOT * Ec);
  const int totalTiles = (Bc * Sc * NSLOT / 16) * (Ec / 16);  // 229376
  out_gemm_kernel<<<totalTiles / 8, 256, 0, stream>>>(Vret, Wout, out);
}